// correlationReverseTransform_38929583571144
// MI455X (gfx1250) — compile-verified
//
#include <hip/hip_runtime.h>

// Problem constants (from the reference):
#define N_ROWS 16384
#define DDIM   2048
#define VDIM   8192

#define ROWS_PER_WG   32            // 2 row-groups of 16
#define LDA           (DDIM + 4)    // padded LDS row stride (floats): bank-conflict free
#define COLS_PER_ITER 128           // 4 col-groups x 2 tiles x 16 cols
#define NUM_ITERS     (VDIM / COLS_PER_ITER)   // 64

typedef __attribute__((ext_vector_type(2))) float v2f;
typedef __attribute__((ext_vector_type(8))) float v8f;

extern "C" __global__ __launch_bounds__(256, 1)
void argmax_gemm_f32_wmma(const float* __restrict__ x,
                          const float* __restrict__ di,
                          float* __restrict__ out)
{
    extern __shared__ float smem[];          // 32 * 2052 * 4B = 256.5 KB (of 320 KB WGP LDS)
    float* a_lds = smem;

    const int tid  = threadIdx.x;
    const int lane = tid & 31;
    const int half = lane >> 4;              // 0: lanes 0-15, 1: lanes 16-31
    const int l16  = lane & 15;
    const int wv   = tid >> 5;               // wave id 0..7
    const int rg   = wv >> 2;                // row group 0..1  (16 rows each)
    const int cg   = wv & 3;                 // col group 0..3  (two 16-col tiles each)
    const int wgRowBase = blockIdx.x * ROWS_PER_WG;

    // ---- Stage this workgroup's 32 x-rows into LDS (b128 loads/stores, padded rows) ----
    for (int i = tid; i < ROWS_PER_WG * (DDIM / 4); i += 256) {
        const int r  = i / (DDIM / 4);
        const int c4 = i - r * (DDIM / 4);
        const float4 v = ((const float4*)(x + (size_t)(wgRowBase + r) * DDIM))[c4];
        *(float4*)&a_lds[r * LDA + c4 * 4] = v;
    }
    __syncthreads();

    // A fragment source: 16x4 fp32 layout -> lanes 0-15 hold K=0,1; lanes 16-31 hold K=2,3
    const float* aRow = &a_lds[(rg * 16 + l16) * LDA + 2 * half];

    float bval[8];
    int   bidx[8];
    #pragma unroll
    for (int r = 0; r < 8; ++r) { bval[r] = -3.402823466e38f; bidx[r] = 0; }

    for (int it = 0; it < NUM_ITERS; ++it) {
        const int col0 = it * COLS_PER_ITER + cg * 16 + l16;   // first 16-col tile
        const int col1 = col0 + 64;                            // second 16-col tile
        // B fragment source: 4x16 fp32 -> lanes 0-15 rows K=0,1; lanes 16-31 rows K=2,3
        const float* bCol0 = di + (size_t)col0 + (size_t)(2 * half) * VDIM;
        const float* bCol1 = bCol0 + 64;

        if (it + 1 < NUM_ITERS)               // warm next column chunk (global_prefetch_b8)
            __builtin_prefetch(di + (size_t)col0 + COLS_PER_ITER, 0, 1);

        // Two independent accumulators -> two independent WMMA dependency chains
        // (fills the matrix pipe at our 2-waves/SIMD occupancy), and each A
        // ds_load is reused by both wmmas.
        v8f acc0 = {};
        v8f acc1 = {};
        #pragma unroll 8
        for (int k = 0; k < DDIM; k += 4) {
            v2f a, b0, b1;
            a.x  = aRow[k];                   // ds_load_b64 (bank-conflict-free via LDA pad)
            a.y  = aRow[k + 1];
            b0.x = bCol0[(size_t)k * VDIM];   // coalesced 64B per half-wave from L2
            b0.y = bCol0[(size_t)(k + 1) * VDIM];
            b1.x = bCol1[(size_t)k * VDIM];
            b1.y = bCol1[(size_t)(k + 1) * VDIM];
            acc0 = __builtin_amdgcn_wmma_f32_16x16x4_f32(
                       false, a, false, b0, (short)0, acc0, false, false);
            acc1 = __builtin_amdgcn_wmma_f32_16x16x4_f32(
                       false, a, false, b1, (short)0, acc1, false, false);
        }

        // Fused running argmax. C layout: VGPR r -> row (half*8 + r), col = l16 slot.
        // Per-lane column visit order is strictly increasing (col0 < col1 < next
        // iteration's col0), so strict '>' keeps the FIRST max like jnp.argmax.
        #pragma unroll
        for (int r = 0; r < 8; ++r) {
            const float v0 = acc0[r];
            if (v0 > bval[r]) { bval[r] = v0; bidx[r] = col0; }
            const float v1 = acc1[r];
            if (v1 > bval[r]) { bval[r] = v1; bidx[r] = col1; }
        }
    }

    // ---- Butterfly reduce across the 16 column lanes (stay within each 16-lane half) ----
    #pragma unroll
    for (int m = 8; m >= 1; m >>= 1) {
        #pragma unroll
        for (int r = 0; r < 8; ++r) {
            const float ov = __shfl_xor(bval[r], m, 32);
            const int   oi = __shfl_xor(bidx[r], m, 32);
            if (ov > bval[r] || (ov == bval[r] && oi < bidx[r])) { bval[r] = ov; bidx[r] = oi; }
        }
    }

    // ---- Cross-wave (4 col-groups) reduce via LDS, then emit indices as float ----
    __syncthreads();                          // a_lds no longer needed; reuse as scratch
    float* sval = smem;                       // [32 rows][4 col-groups]
    int*   sidx = (int*)(smem + ROWS_PER_WG * 4);
    if (l16 == 0) {
        #pragma unroll
        for (int r = 0; r < 8; ++r) {
            const int row = rg * 16 + half * 8 + r;
            sval[row * 4 + cg] = bval[r];
            sidx[row * 4 + cg] = bidx[r];
        }
    }
    __syncthreads();
    if (tid < ROWS_PER_WG) {
        float bv = sval[tid * 4];
        int   bi = sidx[tid * 4];
        #pragma unroll
        for (int c = 1; c < 4; ++c) {
            const float v = sval[tid * 4 + c];
            const int   i = sidx[tid * 4 + c];
            if (v > bv || (v == bv && i < bi)) { bv = v; bi = i; }
        }
        out[wgRowBase + tid] = (float)bi;     // reference stores argmax index as f32
    }
}

extern "C" void kernel_launch(void* const* d_in, const int* in_sizes, int n_in,
                              void* d_out, int out_size, void* d_ws, size_t ws_size,
                              hipStream_t stream) {
    const float* x  = (const float*)d_in[0];   // [N, D] fp32
    const float* di = (const float*)d_in[1];   // [D, V] fp32
    float* out = (float*)d_out;                // [N] fp32 (indices)

    const size_t lds_bytes = (size_t)ROWS_PER_WG * LDA * sizeof(float); // 262,656 B
    hipLaunchKernelGGL(argmax_gemm_f32_wmma,
                       dim3(N_ROWS / ROWS_PER_WG), dim3(256),
                       lds_bytes, stream,
                       x, di, out);
}